// LBPEmbeddingLearned_13675175870631
// MI455X (gfx1250) — compile-verified
//
#include <hip/hip_runtime.h>
#include <hip/hip_bf16.h>

typedef float v2f __attribute__((ext_vector_type(2)));
typedef float v8f __attribute__((ext_vector_type(8)));

#define N_PTS   8192
#define BATCH   4
#define M_ROWS  (N_PTS * BATCH)      // 32768
#define KNN     16
#define D_OUT   288
#define K1      45
#define K1PAD   48
#define BN_EPS  1e-5f

// ---------------------------------------------------------------------------
// Kernel 1: brute-force 16-NN per point + relative-diff feature [M, 48] (pad 3)
// 256 queries per block, candidates tiled through LDS as packed 8:8:8 ints.
// ---------------------------------------------------------------------------
__global__ void __launch_bounds__(256)
knn_diff_kernel(const int* __restrict__ coords, float* __restrict__ diff) {
    const int b = blockIdx.y;
    const int q = blockIdx.x * 256 + threadIdx.x;        // query index in batch
    const int* cb = coords + b * N_PTS * 3;

    const int qx = cb[q * 3 + 0];
    const int qy = cb[q * 3 + 1];
    const int qz = cb[q * 3 + 2];

    __shared__ int spack[256];

    unsigned key[KNN];
    #pragma unroll
    for (int i = 0; i < KNN; ++i) key[i] = 0xFFFFFFFFu;
    unsigned cmax = 0xFFFFFFFFu;
    int maxpos = 0;

    for (int tile = 0; tile < N_PTS; tile += 256) {
        __syncthreads();
        {
            const int idx = tile + threadIdx.x;
            const int x = cb[idx * 3 + 0];
            const int y = cb[idx * 3 + 1];
            const int z = cb[idx * 3 + 2];
            spack[threadIdx.x] = x | (y << 8) | (z << 16);
        }
        __syncthreads();

        #pragma unroll 4
        for (int j = 0; j < 256; ++j) {
            const int p  = spack[j];
            const int dx = qx - (p & 255);
            const int dy = qy - ((p >> 8) & 255);
            const int dz = qz - ((p >> 16) & 255);
            const int d2 = dx * dx + dy * dy + dz * dz;           // <= 195075
            const unsigned kx = ((unsigned)d2 << 13) | (unsigned)(tile + j);
            if (kx < cmax) {                     // replace current max slot
                #pragma unroll
                for (int i = 0; i < KNN; ++i)
                    if (i == maxpos) key[i] = kx;
                cmax = 0u;
                #pragma unroll
                for (int i = 0; i < KNN; ++i)
                    if (key[i] > cmax) { cmax = key[i]; maxpos = i; }
            }
        }
    }

    // full sort (ascending) — constant-index unrolled selection network
    #pragma unroll
    for (int i = 0; i < KNN - 1; ++i)
        #pragma unroll
        for (int j = i + 1; j < KNN; ++j)
            if (key[j] < key[i]) { unsigned t = key[i]; key[i] = key[j]; key[j] = t; }

    const int i0 = (int)(key[0] & 8191u);
    const int px = cb[i0 * 3 + 0];
    const int py = cb[i0 * 3 + 1];
    const int pz = cb[i0 * 3 + 2];

    float* drow = diff + (size_t)(b * N_PTS + q) * K1PAD;
    #pragma unroll
    for (int k = 1; k < KNN; ++k) {
        const int ik = (int)(key[k] & 8191u);
        drow[(k - 1) * 3 + 0] = (float)(cb[ik * 3 + 0] - px);
        drow[(k - 1) * 3 + 1] = (float)(cb[ik * 3 + 1] - py);
        drow[(k - 1) * 3 + 2] = (float)(cb[ik * 3 + 2] - pz);
    }
    drow[45] = 0.0f; drow[46] = 0.0f; drow[47] = 0.0f;
}

// ---------------------------------------------------------------------------
// Kernel 2: h = diff[M,48] @ w1[45,288] + b1   via V_WMMA_F32_16X16X4_F32
// Block = 8 waves, covers 128 rows x 16 cols; B panel staged in LDS (zero-pad).
// ---------------------------------------------------------------------------
__global__ void __launch_bounds__(256)
gemm1_kernel(const float* __restrict__ diff, const float* __restrict__ w1,
             const float* __restrict__ b1, float* __restrict__ h) {
    __shared__ float bs[K1PAD * 16];                  // 3 KB
    const int nb = blockIdx.y * 16;

    for (int i = threadIdx.x; i < K1PAD * 16; i += 256) {
        const int kk = i >> 4, nn = i & 15;
        bs[i] = (kk < K1) ? w1[kk * D_OUT + nb + nn] : 0.0f;
    }
    __syncthreads();

    const int lane  = threadIdx.x & 31;
    const int wave  = threadIdx.x >> 5;
    const int l15   = lane & 15;
    const int hi    = lane >> 4;
    const int mbase = (blockIdx.x * 8 + wave) * 16;
    const int col   = nb + l15;

    const float bias = b1[col];
    v8f c;
    #pragma unroll
    for (int i = 0; i < 8; ++i) c[i] = bias;

    const float* arow = diff + (size_t)(mbase + l15) * K1PAD;

    #pragma unroll
    for (int t = 0; t < K1PAD / 4; ++t) {
        const int kb = 4 * t + hi * 2;
        v2f a;  a[0]  = arow[kb];              a[1]  = arow[kb + 1];
        v2f bm; bm[0] = bs[kb * 16 + l15];     bm[1] = bs[(kb + 1) * 16 + l15];
        c = __builtin_amdgcn_wmma_f32_16x16x4_f32(false, a, false, bm,
                                                  (short)0, c, false, false);
    }

    float* orow = h + (size_t)(mbase + hi * 8) * D_OUT + col;
    #pragma unroll
    for (int j = 0; j < 8; ++j) orow[j * D_OUT] = c[j];
}

// ---------------------------------------------------------------------------
// Kernel 3a: per-column partial sums / sum-of-squares (coalesced rows,
// f32 global atomics into zeroed accumulators).
// ---------------------------------------------------------------------------
__global__ void __launch_bounds__(288)
bn_partial_kernel(const float* __restrict__ h,
                  float* __restrict__ sums, float* __restrict__ sqs) {
    const int c  = threadIdx.x;                       // 0..287
    const int r0 = blockIdx.x * 512;
    float s = 0.0f, qq = 0.0f;
    for (int r = 0; r < 512; ++r) {
        const float v = h[(size_t)(r0 + r) * D_OUT + c];
        s += v; qq += v * v;
    }
    atomicAdd(&sums[c], s);
    atomicAdd(&sqs[c], qq);
}

// Kernel 3b: fold stats into affine scale/shift: y = relu(h*scale + shift)
__global__ void __launch_bounds__(288)
bn_finalize_kernel(const float* __restrict__ sums, const float* __restrict__ sqs,
                   const float* __restrict__ gamma, const float* __restrict__ beta,
                   float* __restrict__ scale, float* __restrict__ shift) {
    const int c = threadIdx.x;
    const float invn = 1.0f / (float)M_ROWS;
    const float mean = sums[c] * invn;
    const float var  = sqs[c] * invn - mean * mean;   // biased, like jnp.var
    const float sc   = gamma[c] * rsqrtf(var + BN_EPS);
    scale[c] = sc;
    shift[c] = beta[c] - mean * sc;
}

// ---------------------------------------------------------------------------
// Kernel 4: out = relu(h*scale+shift) @ w2[288,288] + b2, BN+ReLU fused into
// the WMMA A-fragment load; 288x16 B panel staged in LDS (18 KB).
// ---------------------------------------------------------------------------
__global__ void __launch_bounds__(256)
gemm2_kernel(const float* __restrict__ h,  const float* __restrict__ w2,
             const float* __restrict__ b2, const float* __restrict__ scale,
             const float* __restrict__ shift, float* __restrict__ out) {
    __shared__ float bs[D_OUT * 16];                  // 18 KB
    const int nb = blockIdx.y * 16;

    for (int i = threadIdx.x; i < D_OUT * 16; i += 256) {
        const int kk = i >> 4, nn = i & 15;
        bs[i] = w2[kk * D_OUT + nb + nn];
    }
    __syncthreads();

    const int lane  = threadIdx.x & 31;
    const int wave  = threadIdx.x >> 5;
    const int l15   = lane & 15;
    const int hi    = lane >> 4;
    const int mbase = (blockIdx.x * 8 + wave) * 16;
    const int col   = nb + l15;

    const float bias = b2[col];
    v8f c;
    #pragma unroll
    for (int i = 0; i < 8; ++i) c[i] = bias;

    const float* arow = h + (size_t)(mbase + l15) * D_OUT;
    __builtin_prefetch(arow, 0, 0);                   // global_prefetch_b8
    __builtin_prefetch(arow + 144, 0, 0);

    #pragma unroll 4
    for (int t = 0; t < D_OUT / 4; ++t) {
        const int kb = 4 * t + hi * 2;
        const float a0 = fmaxf(fmaf(arow[kb],     scale[kb],     shift[kb]),     0.0f);
        const float a1 = fmaxf(fmaf(arow[kb + 1], scale[kb + 1], shift[kb + 1]), 0.0f);
        v2f a;  a[0]  = a0;                    a[1]  = a1;
        v2f bm; bm[0] = bs[kb * 16 + l15];     bm[1] = bs[(kb + 1) * 16 + l15];
        c = __builtin_amdgcn_wmma_f32_16x16x4_f32(false, a, false, bm,
                                                  (short)0, c, false, false);
    }

    float* orow = out + (size_t)(mbase + hi * 8) * D_OUT + col;
    #pragma unroll
    for (int j = 0; j < 8; ++j) orow[j * D_OUT] = c[j];
}

// ---------------------------------------------------------------------------
extern "C" void kernel_launch(void* const* d_in, const int* in_sizes, int n_in,
                              void* d_out, int out_size, void* d_ws, size_t ws_size,
                              hipStream_t stream) {
    const int*   coords = (const int*)  d_in[0];   // [4,8192,3] int32
    const float* w1     = (const float*)d_in[1];   // [45,288]
    const float* b1     = (const float*)d_in[2];   // [288]
    const float* gamma  = (const float*)d_in[3];   // [288]
    const float* beta   = (const float*)d_in[4];   // [288]
    const float* w2     = (const float*)d_in[5];   // [288,288]
    const float* b2     = (const float*)d_in[6];   // [288]
    float*       outp   = (float*)d_out;           // [32768,288]

    float* diff = (float*)d_ws;                       // 32768*48
    float* hbuf = diff + (size_t)M_ROWS * K1PAD;      // 32768*288
    float* sums = hbuf + (size_t)M_ROWS * D_OUT;      // 288
    float* sqs  = sums + D_OUT;                       // 288
    float* scal = sqs  + D_OUT;                       // 288
    float* shft = scal + D_OUT;                       // 288

    // zero BN accumulators (ws is poisoned; atomics need clean start each call)
    hipMemsetAsync(sums, 0, 2 * D_OUT * sizeof(float), stream);

    // 1) KNN + diff features
    knn_diff_kernel<<<dim3(N_PTS / 256, BATCH), 256, 0, stream>>>(coords, diff);

    // 2) GEMM1: h = diff @ w1 + b1     (2048 M-tiles / 8 waves, 18 N-tiles)
    gemm1_kernel<<<dim3(M_ROWS / 128, D_OUT / 16), 256, 0, stream>>>(diff, w1, b1, hbuf);

    // 3) BatchNorm statistics -> scale/shift
    bn_partial_kernel<<<dim3(M_ROWS / 512), 288, 0, stream>>>(hbuf, sums, sqs);
    bn_finalize_kernel<<<dim3(1), 288, 0, stream>>>(sums, sqs, gamma, beta, scal, shft);

    // 4) GEMM2 with fused BN+ReLU: out = relu(h*scale+shift) @ w2 + b2
    gemm2_kernel<<<dim3(M_ROWS / 128, D_OUT / 16), 256, 0, stream>>>(
        hbuf, w2, b2, scal, shft, outp);
}